// SparseGCN_81819126989161
// MI455X (gfx1250) — compile-verified
//
#include <hip/hip_runtime.h>

typedef float v2f __attribute__((ext_vector_type(2)));
typedef float v8f __attribute__((ext_vector_type(8)));

#define F1 128   // hidden width (== input feature width)
#define F2 64    // output classes

#define LDA 34   // A-tile pitch in floats (even -> 8B-aligned k-pairs)
#define LDB 34   // B-tile (transposed) pitch in floats

// ---------------------------------------------------------------------------
// Kernel 1: edge scatter  H0[dst] += norm[dst]*norm[src] * X[src]
// wave-per-edge (grid-stride over edges), lane handles 4 contiguous features.
// Gathers are L2-resident (X is 25.6MB << 192MB L2); atomics are HW f32 adds.
// ---------------------------------------------------------------------------
__global__ void gcn_spmm_scatter(const float* __restrict__ x,
                                 const float* __restrict__ norm,
                                 const int* __restrict__ src,
                                 const int* __restrict__ dst,
                                 float* __restrict__ out,
                                 int nEdges)
{
    const int lane   = threadIdx.x & 31;
    const int wave   = blockIdx.x * (blockDim.x >> 5) + (threadIdx.x >> 5);
    const int nWaves = gridDim.x * (blockDim.x >> 5);
    const float4* x4 = (const float4*)x;

    for (int e = wave; e < nEdges; e += nWaves) {
        const int sV = src[e];
        const int dV = dst[e];
        const float c = norm[sV] * norm[dV];
        float4 v = x4[(size_t)sV * (F1 / 4) + lane];
        float* o = out + (size_t)dV * F1 + lane * 4;
        unsafeAtomicAdd(o + 0, c * v.x);
        unsafeAtomicAdd(o + 1, c * v.y);
        unsafeAtomicAdd(o + 2, c * v.z);
        unsafeAtomicAdd(o + 3, c * v.w);
    }
}

// ---------------------------------------------------------------------------
// Kernel 2: H1 = relu(H0 @ W1 + b1) using V_WMMA_F32_16X16X4_F32.
// Block = 256 threads (8 waves), 128 rows per block. K staged in 32-wide
// chunks through LDS. A stored row-major (pitch 34), W1 stored TRANSPOSED
// (pitch 34) so each WMMA operand is one contiguous 8B ds_load_b64 into an
// even-aligned VGPR pair — no register marshalling before v_wmma.
// ---------------------------------------------------------------------------
__global__ void __launch_bounds__(256)
gcn_gemm1_relu(const float* __restrict__ H0,
               const float* __restrict__ W1,
               const float* __restrict__ b1,
               float* __restrict__ H1,
               int nNodes)
{
    __shared__ __align__(16) float lA[128 * LDA];  // [row][k]   128 x 32
    __shared__ __align__(16) float lBt[128 * LDB]; // [n][k]     128 x 32 (W1^T)

    const int tid      = threadIdx.x;
    const int lane     = tid & 31;
    const int wave     = tid >> 5;          // 0..7
    const int half     = lane >> 4;         // 0 or 1
    const int col16    = lane & 15;         // 0..15
    const int blockRow = blockIdx.x * 128;
    const int mlocal   = wave * 16 + col16; // A row this lane reads

    v8f acc[8];
    const v8f z = {0.f, 0.f, 0.f, 0.f, 0.f, 0.f, 0.f, 0.f};
#pragma unroll
    for (int n = 0; n < 8; ++n) acc[n] = z;

    const float4* H04 = (const float4*)H0;
    const float4* W14 = (const float4*)W1;

    for (int kc = 0; kc < 4; ++kc) {        // K chunks of 32
        // --- stage A tile: rows blockRow..+127, cols kc*32..+31 ---
        {
            const int c = tid & 7;          // float4 column within chunk
#pragma unroll
            for (int it = 0; it < 4; ++it) {
                const int row = (tid >> 3) + 32 * it;   // 0..127
                const int gr  = blockRow + row;
                float4 v = {0.f, 0.f, 0.f, 0.f};
                if (gr < nNodes) v = H04[(size_t)gr * 32 + kc * 8 + c];
                float* p = &lA[row * LDA + c * 4];
                p[0] = v.x; p[1] = v.y; p[2] = v.z; p[3] = v.w;
            }
        }
        // --- stage B tile transposed: lBt[n][k] = W1[kc*32+k][n] ---
        {
            const int c = tid & 31;         // float4 column group: n = 4c..4c+3
#pragma unroll
            for (int it = 0; it < 4; ++it) {
                const int kk = (tid >> 5) + 8 * it;     // local k, 0..31
                float4 v = W14[(size_t)(kc * 32 + kk) * 32 + c];
                lBt[(c * 4 + 0) * LDB + kk] = v.x;
                lBt[(c * 4 + 1) * LDB + kk] = v.y;
                lBt[(c * 4 + 2) * LDB + kk] = v.z;
                lBt[(c * 4 + 3) * LDB + kk] = v.w;
            }
        }
        __syncthreads();

        // --- 8 N-tiles x 8 K-steps of WMMA f32 16x16x4 ---
#pragma unroll
        for (int n = 0; n < 8; ++n) {
            const int ncol = n * 16 + col16;
#pragma unroll
            for (int kk = 0; kk < 32; kk += 4) {
                const int ka = kk + 2 * half;           // even -> 8B aligned
                v2f a = *(const v2f*)&lA[mlocal * LDA + ka];
                v2f b = *(const v2f*)&lBt[ncol * LDB + ka];
                acc[n] = __builtin_amdgcn_wmma_f32_16x16x4_f32(
                    false, a, false, b, (short)0, acc[n], false, false);
            }
        }
        __syncthreads();
    }

    // --- epilogue: bias + relu + store (D layout: VGPR r <-> M = r + 8*half)
#pragma unroll
    for (int n = 0; n < 8; ++n) {
        const int col = n * 16 + col16;
        const float bias = b1[col];
#pragma unroll
        for (int r = 0; r < 8; ++r) {
            const int row = blockRow + wave * 16 + r + 8 * half;
            if (row < nNodes) {
                float v = acc[n][r] + bias;
                H1[(size_t)row * F1 + col] = fmaxf(v, 0.f);
            }
        }
    }
}

// ---------------------------------------------------------------------------
// Kernel 3: fused spmm2 + mean:  s[f] = sum_e coef[e] * H1[src[e], f]
// (mean is linear, so mean(spmm(H1)@W2+b2) == (s/N)@W2 + b2 — this removes
// the whole 25.6MB scatter and the [N,64] GEMM.)
// Per-block LDS reduction (ds_add_f32), one global atomic per block.
// ---------------------------------------------------------------------------
__global__ void gcn_edge_colsum(const float* __restrict__ H1,
                                const float* __restrict__ norm,
                                const int* __restrict__ src,
                                const int* __restrict__ dst,
                                float* __restrict__ s,
                                int nEdges)
{
    __shared__ float sblk[F1];
    const int tid = threadIdx.x;
    if (tid < F1) sblk[tid] = 0.f;
    __syncthreads();

    const int lane   = tid & 31;
    const int wave   = blockIdx.x * (blockDim.x >> 5) + (tid >> 5);
    const int nWaves = gridDim.x * (blockDim.x >> 5);
    const float4* h4 = (const float4*)H1;

    float4 acc = {0.f, 0.f, 0.f, 0.f};
    for (int e = wave; e < nEdges; e += nWaves) {
        const int sV = src[e];
        const int dV = dst[e];
        const float c = norm[sV] * norm[dV];
        float4 v = h4[(size_t)sV * (F1 / 4) + lane];
        acc.x += c * v.x;
        acc.y += c * v.y;
        acc.z += c * v.z;
        acc.w += c * v.w;
    }
    atomicAdd(&sblk[lane * 4 + 0], acc.x);
    atomicAdd(&sblk[lane * 4 + 1], acc.y);
    atomicAdd(&sblk[lane * 4 + 2], acc.z);
    atomicAdd(&sblk[lane * 4 + 3], acc.w);
    __syncthreads();
    if (tid < F1) unsafeAtomicAdd(&s[tid], sblk[tid]);
}

// ---------------------------------------------------------------------------
// Kernel 4: out[c] = (s . W2[:,c]) / N + b2[c]   (tiny 128x64 matvec)
// ---------------------------------------------------------------------------
__global__ void gcn_final(const float* __restrict__ s,
                          const float* __restrict__ W2,
                          const float* __restrict__ b2,
                          float* __restrict__ out,
                          float invN)
{
    const int c = threadIdx.x;  // 0..63
    float acc = 0.f;
#pragma unroll 8
    for (int k = 0; k < F1; ++k) acc += s[k] * W2[k * F2 + c];
    out[c] = acc * invN + b2[c];
}

// ---------------------------------------------------------------------------
extern "C" void kernel_launch(void* const* d_in, const int* in_sizes, int n_in,
                              void* d_out, int out_size, void* d_ws, size_t ws_size,
                              hipStream_t stream)
{
    const float* features = (const float*)d_in[0];
    const float* norm     = (const float*)d_in[1];
    const float* W1       = (const float*)d_in[2];
    const float* b1       = (const float*)d_in[3];
    const float* W2       = (const float*)d_in[4];
    const float* b2       = (const float*)d_in[5];
    const int*   src      = (const int*)d_in[6];
    const int*   dst      = (const int*)d_in[7];

    const int nNodes = in_sizes[1];       // norm length
    const int nEdges = in_sizes[6];       // src length

    float* H0   = (float*)d_ws;                        // [N,128]
    float* H1   = H0 + (size_t)nNodes * F1;            // [N,128]
    float* sCol = H1 + (size_t)nNodes * F1;            // [128]

    hipMemsetAsync(H0, 0, (size_t)nNodes * F1 * sizeof(float), stream);
    hipMemsetAsync(sCol, 0, F1 * sizeof(float), stream);

    gcn_spmm_scatter<<<4096, 256, 0, stream>>>(features, norm, src, dst, H0, nEdges);

    const int gBlocks = (nNodes + 127) / 128;
    gcn_gemm1_relu<<<gBlocks, 256, 0, stream>>>(H0, W1, b1, H1, nNodes);

    gcn_edge_colsum<<<1024, 256, 0, stream>>>(H1, norm, src, dst, sCol, nEdges);

    gcn_final<<<1, F2, 0, stream>>>(sCol, W2, b2, (float*)d_out, 1.0f / (float)nNodes);
}